// Yolov2Loss_78597901517407
// MI455X (gfx1250) — compile-verified
//
#include <hip/hip_runtime.h>

// ---------------------------------------------------------------------------
// YOLOv2 loss for MI455X (gfx1250).
// Memory-bound: 148 MB streamed once -> ~6.4 us floor at 23.3 TB/s.
// Strategy: async global->LDS staging (CDNA5 ASYNCcnt path), one thread per
// grid cell, WMMA-based wave32 reduction, one f32 atomic per block.
// ---------------------------------------------------------------------------

#define NCELLS   145                    // (13-1)^2 + 1
#define NBOX     5
#define NGT      16
#define NCLS     20
#define PRED_F   (NCELLS * NBOX * 25)   // 18125 floats per image
#define CELLS_A  73                     // cells handled by first half-block
#define STAGE_MAX (CELLS_A * NBOX * 25) // 9125 floats
#define TPB      128                    // 4 wave32 waves

typedef float v2f __attribute__((ext_vector_type(2)));
typedef float v8f __attribute__((ext_vector_type(8)));

typedef __attribute__((address_space(1))) int* gptr_t;  // global (device) int*
typedef __attribute__((address_space(3))) int* lptr_t;  // LDS (shared) int*

#ifdef __has_builtin
#  if __has_builtin(__builtin_amdgcn_global_load_async_to_lds_b32)
#    define HAVE_ASYNC 1
#  endif
#  if __has_builtin(__builtin_amdgcn_s_wait_asynccnt)
#    define HAVE_WAITA 1
#  endif
#endif
#ifndef HAVE_ASYNC
#  define HAVE_ASYNC 0
#endif
#ifndef HAVE_WAITA
#  define HAVE_WAITA 0
#endif

__global__ void __launch_bounds__(64) yolo_zero_kernel(float* p) { *p = 0.0f; }

__global__ void __launch_bounds__(TPB) yolo_loss_kernel(
    const float* __restrict__ gt, const float* __restrict__ out,
    float* __restrict__ loss) {
  extern __shared__ float smem[];
  float* pred_s = smem;               // STAGE_MAX floats
  float* gt_s   = smem + STAGE_MAX;   // 16*5
  float* garea  = gt_s + NGT * 5;     // 16
  float* wsum   = garea + NGT;        // 4 (one per wave)

  const int tid    = threadIdx.x;
  const int img    = blockIdx.x >> 1;
  const int half   = blockIdx.x & 1;
  const int cbase  = half * CELLS_A;
  const int ccount = half ? (NCELLS - CELLS_A) : CELLS_A;
  const int nstage = ccount * NBOX * 25;

  const float* src  = out + (size_t)img * PRED_F + (size_t)cbase * (NBOX * 25);
  const float* gsrc = gt + (size_t)img * (NGT * 5);

  // ---- stage prediction slab + GT into LDS (async on CDNA5) ----
  for (int i = tid; i < nstage; i += TPB) {
#if HAVE_ASYNC
    __builtin_amdgcn_global_load_async_to_lds_b32(
        (gptr_t)(src + i), (lptr_t)(pred_s + i), 0, 0);
#else
    pred_s[i] = src[i];
#endif
  }
  if (tid < NGT * 5) {
#if HAVE_ASYNC
    __builtin_amdgcn_global_load_async_to_lds_b32(
        (gptr_t)(gsrc + tid), (lptr_t)(gt_s + tid), 0, 0);
#else
    gt_s[tid] = gsrc[tid];
#endif
  }
#if HAVE_ASYNC
#  if HAVE_WAITA
  __builtin_amdgcn_s_wait_asynccnt(0);
#  else
  asm volatile("s_wait_asynccnt 0" ::: "memory");
#  endif
#endif
  __syncthreads();

  if (tid < NGT) {
    float x1 = gt_s[tid * 5 + 0], y1 = gt_s[tid * 5 + 1];
    float x2 = gt_s[tid * 5 + 2], y2 = gt_s[tid * 5 + 3];
    garea[tid] = fmaxf(y2 - y1, 0.0f) * fmaxf(x2 - x1, 0.0f);
  }
  __syncthreads();

  float partial = 0.0f;
  if (tid < ccount) {
    const float* P = pred_s + tid * (NBOX * 25);

    float chat[NBOX];
    float conf[NBOX];
    float maxv = -1.0f;   // iou >= 0, so strict '>' keeps first-occurrence argmax
    int   argf = 0;

#pragma unroll
    for (int bb = 0; bb < NBOX; ++bb) {
      const float a0 = P[bb * 25 + 0], a1 = P[bb * 25 + 1];
      const float a2 = P[bb * 25 + 2], a3 = P[bb * 25 + 3];
      conf[bb] = P[bb * 25 + 4];
      const float areaA = fmaxf(a3 - a1, 0.0f) * fmaxf(a2 - a0, 0.0f);
      float cmax = 0.0f;
      for (int g = 0; g < NGT; ++g) {
        const float b0 = gt_s[g * 5 + 0], b1 = gt_s[g * 5 + 1];
        const float b2 = gt_s[g * 5 + 2], b3 = gt_s[g * 5 + 3];
        const float xi1 = fmaxf(a0, b0), yi1 = fmaxf(a1, b1);
        const float xi2 = fminf(a2, b2), yi2 = fminf(a3, b3);
        const float inter = fmaxf(yi2 - yi1, 0.0f) * fmaxf(xi2 - xi1, 0.0f);
        const float uni = areaA + garea[g] - inter;
        const float iou = (uni > 0.0f) ? (inter / uni) : 0.0f;
        cmax = fmaxf(cmax, iou);
        if (iou > maxv) { maxv = iou; argf = bb * NGT + g; }
      }
      chat[bb] = cmax;
    }

    const int ind_bb = argf / NGT;
    const int ind_gt = argf - ind_bb * NGT;

    // ---- localization loss ----
    float loc = 0.0f;
    if (maxv > 0.0f) {
      const float bx = P[ind_bb * 25 + 0] * 416.0f;
      const float by = P[ind_bb * 25 + 1] * 416.0f;
      const float bw = P[ind_bb * 25 + 2] * 416.0f;
      const float bh = P[ind_bb * 25 + 3] * 416.0f;
      const float g0 = gt_s[ind_gt * 5 + 0], g1 = gt_s[ind_gt * 5 + 1];
      const float g2 = gt_s[ind_gt * 5 + 2], g3 = gt_s[ind_gt * 5 + 3];
      const float dx = bx - g0, dy = by - g1;
      const float dw = sqrtf(bw) - sqrtf(g2);
      const float dh = sqrtf(bh) - sqrtf(g3);
      loc = 5.0f * (dx * dx + dy * dy + dw * dw + dh * dh);
    }

    // ---- confidence loss ----
    const int ind_max_bb = (maxv > 0.0f) ? ind_bb : (NBOX - 1);
    float conf_loss = 0.0f;
#pragma unroll
    for (int bb = 0; bb < NBOX; ++bb) {
      float d = chat[bb] * (conf[bb] - 1.0f);
      d = d * d;
      conf_loss += ((bb == ind_max_bb) ? 1.0f : 0.5f) * d;
    }

    // ---- class loss ----
    float maxobj = -1e30f;
    int   ind_bbox = 0;
#pragma unroll
    for (int bb = 0; bb < NBOX; ++bb)
      if (conf[bb] > maxobj) { maxobj = conf[bb]; ind_bbox = bb; }

    float cls_loss = 0.0f;
    if (maxobj >= 0.6f) {
      const float a0 = P[ind_bbox * 25 + 0], a1 = P[ind_bbox * 25 + 1];
      const float a2 = P[ind_bbox * 25 + 2], a3 = P[ind_bbox * 25 + 3];
      const float areaA = fmaxf(a3 - a1, 0.0f) * fmaxf(a2 - a0, 0.0f);
      float clsmax = -1.0f;
      int   clsarg = 0;
      for (int g = 0; g < NGT; ++g) {
        const float b0 = gt_s[g * 5 + 0], b1 = gt_s[g * 5 + 1];
        const float b2 = gt_s[g * 5 + 2], b3 = gt_s[g * 5 + 3];
        const float xi1 = fmaxf(a0, b0), yi1 = fmaxf(a1, b1);
        const float xi2 = fminf(a2, b2), yi2 = fminf(a3, b3);
        const float inter = fmaxf(yi2 - yi1, 0.0f) * fmaxf(xi2 - xi1, 0.0f);
        const float uni = areaA + garea[g] - inter;
        const float iou = (uni > 0.0f) ? (inter / uni) : 0.0f;
        if (iou > clsmax) { clsmax = iou; clsarg = g; }
      }
      const int class_gt = (clsmax > 0.0f) ? (int)gt_s[clsarg * 5 + 4] : (NCLS - 1);
      for (int k = 0; k < NCLS; ++k) {
        const float p = P[ind_bbox * 25 + 5 + k] * maxobj;  // conf[ind_bbox] == maxobj
        const float t = p - ((k == class_gt) ? 1.0f : 0.0f);
        cls_loss += t * t;
      }
    }
    partial = loc + conf_loss + cls_loss;
  }

  // ---- wave32 reduction via v_wmma_f32_16x16x4_f32 ----
  // A (16x4): lane L<16 -> row L gets {p,0} at K=0,1; lane L>=16 -> row L-16
  // gets {p,0} at K=2,3.  With B = all-ones, D[m][*] = p[m] + p[m+16].
  // EXEC is all-ones here (no divergence at this point; block is 4 full waves).
  v2f A;  A.x = partial; A.y = 0.0f;
  v2f Bv; Bv.x = 1.0f;  Bv.y = 1.0f;
  v8f C = {};
  v8f D = __builtin_amdgcn_wmma_f32_16x16x4_f32(
      /*neg_a=*/false, A, /*neg_b=*/false, Bv,
      /*c_mod=*/(short)0, C, /*reuse_a=*/false, /*reuse_b=*/false);
  float s = D[0] + D[1] + D[2] + D[3] + D[4] + D[5] + D[6] + D[7];
  // lanes 0-15 hold sum of rows 0-7, lanes 16-31 hold sum of rows 8-15.
  s += __shfl_xor(s, 16);

  const int lane = tid & 31;
  const int wave = tid >> 5;
  if (lane == 0) wsum[wave] = s;
  __syncthreads();
  if (tid == 0) {
    float tot = wsum[0] + wsum[1] + wsum[2] + wsum[3];
    atomicAdd(loss, tot);
  }
}

extern "C" void kernel_launch(void* const* d_in, const int* in_sizes, int n_in,
                              void* d_out, int out_size, void* d_ws, size_t ws_size,
                              hipStream_t stream) {
  (void)n_in; (void)out_size; (void)d_ws; (void)ws_size;
  const float* gt  = (const float*)d_in[0];   // (B, 16, 5)
  const float* out = (const float*)d_in[1];   // (B, 725, 25)
  float* loss = (float*)d_out;                // scalar f32

  const int batch = in_sizes[0] / (NGT * 5);  // 2048

  yolo_zero_kernel<<<dim3(1), dim3(64), 0, stream>>>(loss);

  const size_t shbytes = (size_t)(STAGE_MAX + NGT * 5 + NGT + 4) * sizeof(float);
  yolo_loss_kernel<<<dim3(batch * 2), dim3(TPB), shbytes, stream>>>(gt, out, loss);
}